// Kmeans_70789650972754
// MI455X (gfx1250) — compile-verified
//
#include <hip/hip_runtime.h>
#include <hip/hip_bf16.h>

typedef __attribute__((ext_vector_type(2))) float v2f;
typedef __attribute__((ext_vector_type(4))) float v4f;
typedef __attribute__((ext_vector_type(8))) float v8f;

#define K_CENT      512
#define DIM         64
#define LDS_STRIDE  68                       // padded row stride in floats (272B, 16B aligned)
#define WAVES_PER_WG 8
#define ROWS_PER_WAVE 16
#define ROWS_PER_WG (WAVES_PER_WG * ROWS_PER_WAVE)   // 128
#define NTILES      (K_CENT / 16)            // 32 centroid tiles

__global__ void __launch_bounds__(256, 1)
kmeans_assign_wmma(const float* __restrict__ X,
                   const float* __restrict__ C,
                   int* __restrict__ out,
                   int N)
{
    extern __shared__ float lds[];
    float* cl = lds;                         // [512][68] padded centroid rows
    float* c2 = lds + K_CENT * LDS_STRIDE;   // [512] squared norms

    const int tid = threadIdx.x;

    // ---- Stage centroids into LDS (float4 coalesced; padded stride keeps 16B align) ----
    {
        const v4f* C4 = (const v4f*)C;
        #pragma unroll
        for (int i = 0; i < (K_CENT * DIM / 4) / 256; ++i) {   // 32 iters
            int idx = i * 256 + tid;         // f4 index
            int k   = idx >> 4;              // 16 f4 per centroid row
            int d4  = idx & 15;
            *(v4f*)&cl[k * LDS_STRIDE + d4 * 4] = C4[idx];
        }
    }
    __syncthreads();

    // ---- Precompute ||c||^2 ----
    for (int k = tid; k < K_CENT; k += 256) {
        const float* row = &cl[k * LDS_STRIDE];
        float s = 0.f;
        #pragma unroll
        for (int d = 0; d < DIM; ++d) s = fmaf(row[d], row[d], s);
        c2[k] = s;
    }
    __syncthreads();

    // ---- Per-wave 16-row GEMM tile ----
    const int wave  = tid >> 5;
    const int lane  = tid & 31;
    const int laneM = lane & 15;             // row-in-tile for A, column n for B/D
    const int hi    = lane >> 4;             // 0: K(dim) 0..1 half, 1: K(dim) 2..3 half

    const int rowBase = blockIdx.x * ROWS_PER_WG + wave * ROWS_PER_WAVE;
    int row  = rowBase + laneM;
    int rowc = row < N ? row : (N - 1);      // clamp tail loads (stores are guarded)

    // A fragments: frag t covers dim 4t..4t+3; this lane holds cols 4t+2*hi, 4t+2*hi+1
    v2f a[16];
    {
        const float* xrow = X + (size_t)rowc * DIM;
        #pragma unroll
        for (int t = 0; t < 16; ++t)
            a[t] = *(const v2f*)(xrow + 4 * t + 2 * hi);
    }

    float minv[8];
    int   mini[8];
    #pragma unroll
    for (int r = 0; r < 8; ++r) { minv[r] = 3.402823466e38f; mini[r] = 0; }

    for (int kt = 0; kt < NTILES; ++kt) {
        // B fragment base: B[k][n] = C[kt*16 + n][k]; this lane serves column n = laneM
        const float* bbase = &cl[(kt * 16 + laneM) * LDS_STRIDE + 2 * hi];

        v8f acc0 = {};
        v8f acc1 = {};
        #pragma unroll
        for (int t = 0; t < 16; t += 2) {
            v2f b0 = *(const v2f*)(bbase + 4 * t);
            v2f b1 = *(const v2f*)(bbase + 4 * (t + 1));
            acc0 = __builtin_amdgcn_wmma_f32_16x16x4_f32(
                       false, a[t],     false, b0, (short)0, acc0, false, false);
            acc1 = __builtin_amdgcn_wmma_f32_16x16x4_f32(
                       false, a[t + 1], false, b1, (short)0, acc1, false, false);
        }

        const int   nidx = kt * 16 + laneM;
        const float cn   = c2[nidx];
        #pragma unroll
        for (int r = 0; r < 8; ++r) {
            float dot = acc0[r] + acc1[r];
            float s   = fmaf(-2.0f, dot, cn);        // ||c||^2 - 2 x.c
            if (s < minv[r]) { minv[r] = s; mini[r] = nidx; }
        }
    }

    // ---- Cross-lane argmin reduction within each 16-lane half (xor 1,2,4,8) ----
    #pragma unroll
    for (int m = 1; m < 16; m <<= 1) {
        #pragma unroll
        for (int r = 0; r < 8; ++r) {
            float ov = __shfl_xor(minv[r], m, 32);
            int   oi = __shfl_xor(mini[r], m, 32);
            if (ov < minv[r] || (ov == minv[r] && oi < mini[r])) {
                minv[r] = ov; mini[r] = oi;
            }
        }
    }

    // Lane 0 holds rows rowBase+0..7, lane 16 holds rows rowBase+8..15
    if (laneM == 0) {
        const int rbase = rowBase + hi * 8;
        #pragma unroll
        for (int r = 0; r < 8; ++r) {
            int rr = rbase + r;
            if (rr < N) out[rr] = mini[r];
        }
    }
}

extern "C" void kernel_launch(void* const* d_in, const int* in_sizes, int n_in,
                              void* d_out, int out_size, void* d_ws, size_t ws_size,
                              hipStream_t stream) {
    const float* X = (const float*)d_in[0];      // [N, 64] f32
    const float* C = (const float*)d_in[1];      // [512, 64] f32
    int* out = (int*)d_out;                      // [N] int32 argmin indices

    const int N = in_sizes[0] / DIM;
    const int grid = (N + ROWS_PER_WG - 1) / ROWS_PER_WG;
    const size_t shmem = (size_t)(K_CENT * LDS_STRIDE + K_CENT) * sizeof(float); // ~138 KB

    kmeans_assign_wmma<<<grid, 256, shmem, stream>>>(X, C, out, N);
}